// UHGSAGEConv_78357383348672
// MI455X (gfx1250) — compile-verified
//
#include <hip/hip_runtime.h>
#include <math.h>

#define EPSF 1e-8f

typedef __attribute__((ext_vector_type(2))) float v2f;
typedef __attribute__((ext_vector_type(8))) float v8f;

__device__ __forceinline__ float wave_sum32(float v) {
#pragma unroll
    for (int off = 16; off > 0; off >>= 1)
        v += __shfl_xor(v, off, 32);
    return v;
}

// ---------------------------------------------------------------------------
// K1: xn = normalize_points(x).  One wave per row (wave32).
// ---------------------------------------------------------------------------
__global__ void k_normalize_x(const float* __restrict__ x,
                              float* __restrict__ xn, int N) {
    int wave = (blockIdx.x * blockDim.x + threadIdx.x) >> 5;
    int lane = threadIdx.x & 31;
    if (wave >= N) return;
    const float* xr = x + (size_t)wave * 257;
    float f[8];
    float ss = 0.f;
#pragma unroll
    for (int j = 0; j < 8; ++j) {
        f[j] = xr[lane + 32 * j];
        ss += f[j] * f[j];
    }
    ss = wave_sum32(ss);
    float h = xr[256];
    float s = (h < 0.f) ? -1.f : 1.f;   // sign(0) -> 1 per reference
    float* orow = xn + (size_t)wave * 257;
    if (ss == 0.f) {                    // zero_mask: f -> ones, norm = 16
        float v = s * (1.f / 16.f);
#pragma unroll
        for (int j = 0; j < 8; ++j) orow[lane + 32 * j] = v;
    } else {
        float inv = s / fmaxf(sqrtf(ss), EPSF);
#pragma unroll
        for (int j = 0; j < 8; ++j) orow[lane + 32 * j] = f[j] * inv;
    }
    if (lane == 0) orow[256] = h * s;
}

// ---------------------------------------------------------------------------
// K2: fp32 WMMA GEMM   Y = Xn_f (N x 256) @ W^T (256 x 256)
// One wave computes a 16x64 output tile (4 accumulators), so each A fragment
// load feeds 4 v_wmma_f32_16x16x4_f32 ops (1.25 loads per WMMA instead of 2,
// and 4x less L2 traffic on the 20.5 MB activation matrix).
// gridDim.y: 0 -> W_self/self_full, 1 -> W_neigh/neigh_full.
// Output buffers have pitch 257 (h column filled later).
// ---------------------------------------------------------------------------
__global__ void k_gemm_wmma(const float* __restrict__ xn,
                            const float* __restrict__ Wself,
                            const float* __restrict__ Wneigh,
                            float* __restrict__ self_full,
                            float* __restrict__ neigh_full, int N) {
    const int NGROUPS = 4;              // 256 cols / 64 cols-per-wave
    int wave = (blockIdx.x * blockDim.x + threadIdx.x) >> 5;
    int lane = threadIdx.x & 31;
    int ngrp  = wave % NGROUPS;
    int mtile = wave / NGROUPS;
    int m0 = mtile * 16;
    if (m0 >= N) return;

    const float* W = (blockIdx.y == 0) ? Wself : Wneigh;
    float* Y       = (blockIdx.y == 0) ? self_full : neigh_full;
    int n0 = ngrp * 64;

    // A fragment (16x4 f32): lane<16 -> row m0+lane, K = {k0,k0+1};
    //                        lane>=16 -> same rows, K = {k0+2,k0+3}.
    int mrow  = m0 + (lane & 15);
    if (mrow >= N) mrow = N - 1;        // padded rows: computed, never stored
    int khalf = (lane >> 4) * 2;
    // B fragment (4x16 f32): lane selects column n, same K split.
    // B[k][n] = W[n*256 + k]  (B = W^T), contiguous float2 per lane.
    int ncol  = n0 + (lane & 15);

    const float* ap  = xn + (size_t)mrow * 257 + khalf;
    const float* bp0 = W + (size_t)(ncol +  0) * 256 + khalf;
    const float* bp1 = W + (size_t)(ncol + 16) * 256 + khalf;
    const float* bp2 = W + (size_t)(ncol + 32) * 256 + khalf;
    const float* bp3 = W + (size_t)(ncol + 48) * 256 + khalf;

    v8f c0 = {0.f, 0.f, 0.f, 0.f, 0.f, 0.f, 0.f, 0.f};
    v8f c1 = c0, c2 = c0, c3 = c0;
#pragma unroll 2
    for (int k0 = 0; k0 < 256; k0 += 4) {
        v2f a, b0, b1, b2, b3;
        a.x  = ap[k0];   a.y  = ap[k0 + 1];
        b0.x = bp0[k0];  b0.y = bp0[k0 + 1];
        b1.x = bp1[k0];  b1.y = bp1[k0 + 1];
        b2.x = bp2[k0];  b2.y = bp2[k0 + 1];
        b3.x = bp3[k0];  b3.y = bp3[k0 + 1];
        // 8 args: (neg_a, A, neg_b, B, c_mod, C, reuse_a, reuse_b)
        c0 = __builtin_amdgcn_wmma_f32_16x16x4_f32(false, a, false, b0,
                                                   (short)0, c0, false, false);
        c1 = __builtin_amdgcn_wmma_f32_16x16x4_f32(false, a, false, b1,
                                                   (short)0, c1, false, false);
        c2 = __builtin_amdgcn_wmma_f32_16x16x4_f32(false, a, false, b2,
                                                   (short)0, c2, false, false);
        c3 = __builtin_amdgcn_wmma_f32_16x16x4_f32(false, a, false, b3,
                                                   (short)0, c3, false, false);
    }

    // D layout: VGPR v, lane l -> M = m0 + v + (l>>4)*8, N = n0 + (l&15)
    int mbase = m0 + (lane >> 4) * 8;
    int nout  = n0 + (lane & 15);
    if (m0 + 16 <= N) {                 // whole tile in range: straight stores
#pragma unroll
        for (int v = 0; v < 8; ++v) {
            float* yr = Y + (size_t)(mbase + v) * 257 + nout;
            yr[0]  = c0[v];
            yr[16] = c1[v];
            yr[32] = c2[v];
            yr[48] = c3[v];
        }
    } else {
#pragma unroll
        for (int v = 0; v < 8; ++v) {
            int m = mbase + v;
            if (m < N) {
                float* yr = Y + (size_t)m * 257 + nout;
                yr[0]  = c0[v];
                yr[16] = c1[v];
                yr[32] = c2[v];
                yr[48] = c3[v];
            }
        }
    }
}

// ---------------------------------------------------------------------------
// K3: finish uhg_transform: normalize t rows in place, append h (>=0 already,
// so the outer normalize_points sign is +1 and is idempotent on unit vectors).
// ---------------------------------------------------------------------------
__global__ void k_row_finish(float* __restrict__ Y,
                             const float* __restrict__ xn, int N) {
    int wave = (blockIdx.x * blockDim.x + threadIdx.x) >> 5;
    int lane = threadIdx.x & 31;
    if (wave >= N) return;
    float* yr = Y + (size_t)wave * 257;
    float t[8];
    float ss = 0.f;
#pragma unroll
    for (int j = 0; j < 8; ++j) {
        t[j] = yr[lane + 32 * j];
        ss += t[j] * t[j];
    }
    ss = wave_sum32(ss);
    if (ss == 0.f) {                    // t/EPS == 0 -> zero_mask path -> ones/16
        float v = 1.f / 16.f;
#pragma unroll
        for (int j = 0; j < 8; ++j) yr[lane + 32 * j] = v;
    } else {
        float inv = 1.f / fmaxf(sqrtf(ss), EPSF);
#pragma unroll
        for (int j = 0; j < 8; ++j) yr[lane + 32 * j] = t[j] * inv;
    }
    if (lane == 0) yr[256] = xn[(size_t)wave * 257 + 256];  // h >= 0
}

// ---------------------------------------------------------------------------
// K4a: init accumulator (h column = 1.0) and counts. Deterministic each call.
// ---------------------------------------------------------------------------
__global__ void k_init_acc(float* __restrict__ acc, float* __restrict__ cnt,
                           int N) {
    int i = blockIdx.x * blockDim.x + threadIdx.x;
    int total = N * 257;
    if (i < total) acc[i] = ((i % 257) == 256) ? 1.0f : 0.0f;
    if (i < N) cnt[i] = 0.f;
}

// ---------------------------------------------------------------------------
// K4b: edge scatter-add. One wave per edge: gather neigh_full[col] (L2-hot),
// atomic-add into acc[row]. Lane 0 bumps the degree count.
// ---------------------------------------------------------------------------
__global__ void k_scatter(const int* __restrict__ ei,
                          const float* __restrict__ nf,
                          float* __restrict__ acc, float* __restrict__ cnt,
                          int E) {
    int wave = (blockIdx.x * blockDim.x + threadIdx.x) >> 5;
    int lane = threadIdx.x & 31;
    if (wave >= E) return;
    int r = ei[wave];          // row = edge_index[0][e]
    int c = ei[E + wave];      // col = edge_index[1][e]
    const float* src = nf + (size_t)c * 257;
    float* dst = acc + (size_t)r * 257;
#pragma unroll
    for (int j = 0; j < 8; ++j)
        atomicAdd(&dst[lane + 32 * j], src[lane + 32 * j]);
    if (lane == 0) {
        atomicAdd(&dst[256], src[256]);
        atomicAdd(&cnt[r], 1.0f);
    }
}

// ---------------------------------------------------------------------------
// K5: finalize: mean, normalize(out) -> p2, avg with p1, normalize twice.
// ---------------------------------------------------------------------------
__global__ void k_finalize(const float* __restrict__ self_full,
                           const float* __restrict__ acc,
                           const float* __restrict__ cnt,
                           float* __restrict__ out, int N) {
    int wave = (blockIdx.x * blockDim.x + threadIdx.x) >> 5;
    int lane = threadIdx.x & 31;
    if (wave >= N) return;
    const float* p1 = self_full + (size_t)wave * 257;
    const float* ar = acc + (size_t)wave * 257;
    float c = fmaxf(cnt[wave], 1.0f);
    float invc = 1.0f / c;

    // p2 = normalize_points([acc_f / c, acc_h])
    float f2[8];
    float ss = 0.f;
#pragma unroll
    for (int j = 0; j < 8; ++j) {
        f2[j] = ar[lane + 32 * j] * invc;
        ss += f2[j] * f2[j];
    }
    ss = wave_sum32(ss);
    float h2 = ar[256];
    float s2 = (h2 < 0.f) ? -1.f : 1.f;
    if (ss == 0.f) {
#pragma unroll
        for (int j = 0; j < 8; ++j) f2[j] = s2 * (1.f / 16.f);
    } else {
        float inv = s2 / fmaxf(sqrtf(ss), EPSF);
#pragma unroll
        for (int j = 0; j < 8; ++j) f2[j] *= inv;
    }
    float p2h = h2 * s2;

    // avg = (p1*t + p2) / max(t+1, EPS);  t = clip(0.5/(0.5+EPS), EPS, 1e8)
    const float t = 0.5f / (1.0f - 0.5f + 1e-8f);
    float invt1 = 1.0f / fmaxf(t + 1.0f, EPSF);
    float a[8];
    float ss2 = 0.f;
#pragma unroll
    for (int j = 0; j < 8; ++j) {
        a[j] = (p1[lane + 32 * j] * t + f2[j]) * invt1;
        ss2 += a[j] * a[j];
    }
    float ah = (p1[256] * t + p2h) * invt1;
    ss2 = wave_sum32(ss2);

    // combined = normalize_points(avg)
    float sc = (ah < 0.f) ? -1.f : 1.f;
    float g[8];
    if (ss2 == 0.f) {
#pragma unroll
        for (int j = 0; j < 8; ++j) g[j] = sc * (1.f / 16.f);
    } else {
        float inv = sc / fmaxf(sqrtf(ss2), EPSF);
#pragma unroll
        for (int j = 0; j < 8; ++j) g[j] = a[j] * inv;
    }
    float gh = ah * sc;

    // final normalize_points(combined) -- exact replay for bit fidelity
    float ss3 = 0.f;
#pragma unroll
    for (int j = 0; j < 8; ++j) ss3 += g[j] * g[j];
    ss3 = wave_sum32(ss3);
    float s3 = (gh < 0.f) ? -1.f : 1.f;
    float* o = out + (size_t)wave * 257;
    if (ss3 == 0.f) {
#pragma unroll
        for (int j = 0; j < 8; ++j) o[lane + 32 * j] = s3 * (1.f / 16.f);
    } else {
        float inv = s3 / fmaxf(sqrtf(ss3), EPSF);
#pragma unroll
        for (int j = 0; j < 8; ++j) o[lane + 32 * j] = g[j] * inv;
    }
    if (lane == 0) o[256] = gh * s3;
}

// ---------------------------------------------------------------------------
extern "C" void kernel_launch(void* const* d_in, const int* in_sizes, int n_in,
                              void* d_out, int out_size, void* d_ws,
                              size_t ws_size, hipStream_t stream) {
    const float* x      = (const float*)d_in[0];   // (N, 257)
    const float* Wself  = (const float*)d_in[1];   // (256, 256)
    const float* Wneigh = (const float*)d_in[2];   // (256, 256)
    const int*   eidx   = (const int*)d_in[3];     // (2, E)

    const int N = in_sizes[0] / 257;
    const int E = in_sizes[3] / 2;

    // workspace layout (floats)
    float* ws         = (float*)d_ws;
    float* xn         = ws;                          // N*257
    float* self_full  = xn + (size_t)N * 257;        // N*257 (GEMM out, pitch 257)
    float* neigh_full = self_full + (size_t)N * 257; // N*257
    float* acc        = neigh_full + (size_t)N * 257;// N*257
    float* cnt        = acc + (size_t)N * 257;       // N

    const int WPB = 8;                 // waves per 256-thread block
    int rowBlocks = (N + WPB - 1) / WPB;

    // K1: normalize input rows
    k_normalize_x<<<rowBlocks, 256, 0, stream>>>(x, xn, N);

    // K2: both GEMMs via fp32 WMMA (grid.y = which weight);
    //     one wave per 16x64 output tile -> mtiles * 4 waves.
    int mtiles = (N + 15) / 16;
    int totalWaves = mtiles * 4;
    dim3 ggrid((totalWaves + WPB - 1) / WPB, 2, 1);
    k_gemm_wmma<<<ggrid, 256, 0, stream>>>(xn, Wself, Wneigh, self_full,
                                           neigh_full, N);

    // K3: normalize transform rows, attach h
    k_row_finish<<<rowBlocks, 256, 0, stream>>>(self_full, xn, N);
    k_row_finish<<<rowBlocks, 256, 0, stream>>>(neigh_full, xn, N);

    // K4a: init accumulators (fresh every call — graph-replay safe)
    int initN = N * 257;
    k_init_acc<<<(initN + 255) / 256, 256, 0, stream>>>(acc, cnt, N);

    // K4b: edge scatter-add (wave per edge)
    int edgeBlocks = (E + WPB - 1) / WPB;
    k_scatter<<<edgeBlocks, 256, 0, stream>>>(eidx, neigh_full, acc, cnt, E);

    // K5: mean + normalize chain -> output
    k_finalize<<<rowBlocks, 256, 0, stream>>>(self_full, acc, cnt,
                                              (float*)d_out, N);
}